// attention_score_85693187489968
// MI455X (gfx1250) — compile-verified
//
#include <hip/hip_runtime.h>

// GCNConv (out_channels=1) for MI455X / gfx1250, wave32.
//
// Pipeline (5 kernels, all on `stream`, workspace-carved scratch):
//   1. init:     deg[i]=1 (self loop), acc[i]=0
//   2. degree:   atomicAdd(deg[col[e]], 1)            (int atomics, L2-resident)
//   3. xw_wmma:  h = x@W via V_WMMA_F32_16X16X4_F32; fuse dinv=rsqrt(deg),
//                hs = dinv*h   (only HBM-heavy kernel: streams 102.4 MB of x)
//   4. scatter:  acc[col[e]] += hs[row[e]]            (native f32 atomics, L2)
//   5. finalize: out[i] = dinv[i]*(acc[i]+hs[i]) + b
//
// Bandwidth floor ~155 MB / 23.3 TB/s ~= 6.7 us; all per-node arrays (~400 KB)
// are L2-resident, so edge gathers/scatters never touch HBM after first use.

#define D_FEAT 256
#define ROW_STRIDE 260          // 256 + 4 pad; 260 % 64 == 4 -> conflict-free b64 A-tile loads
#define NODES_PER_WAVE 16
#define WAVES_PER_BLOCK 2
#define NODES_PER_BLOCK (NODES_PER_WAVE * WAVES_PER_BLOCK)

typedef float v2f __attribute__((ext_vector_type(2)));
typedef float v8f __attribute__((ext_vector_type(8)));

__global__ void gcn_init_kernel(int* __restrict__ deg, float* __restrict__ acc, int n) {
    int i = blockIdx.x * blockDim.x + threadIdx.x;
    if (i < n) { deg[i] = 1; acc[i] = 0.0f; }
}

// 4 edges per thread: one global_load_b128 on the col stream, then 4 L2 atomics.
__global__ void gcn_degree_kernel(const int* __restrict__ col, int* __restrict__ deg, int e) {
    int i4 = (blockIdx.x * blockDim.x + threadIdx.x) * 4;
    if (i4 + 3 < e) {
        int4 c = *(const int4*)(col + i4);
        atomicAdd(&deg[c.x], 1);
        atomicAdd(&deg[c.y], 1);
        atomicAdd(&deg[c.z], 1);
        atomicAdd(&deg[c.w], 1);
    } else {
        for (int i = i4; i < e; ++i) atomicAdd(&deg[col[i]], 1);
    }
}

// One wave -> 16 nodes. A = x tile (16x4 fp32), B = W chunk replicated across
// 16 columns, C accumulates over K=256 in 64 WMMA steps.
__global__ __launch_bounds__(WAVES_PER_BLOCK * 32)
void gcn_xw_wmma_kernel(const float* __restrict__ x, const float* __restrict__ w,
                        const int* __restrict__ deg, float* __restrict__ dinv,
                        float* __restrict__ hs, int n) {
    __shared__ float lds[WAVES_PER_BLOCK * NODES_PER_WAVE * ROW_STRIDE + D_FEAT];
    float* wlds = &lds[WAVES_PER_BLOCK * NODES_PER_WAVE * ROW_STRIDE];

    const int tid  = threadIdx.x;
    const int wave = tid >> 5;
    const int lane = tid & 31;
    const int node_base = blockIdx.x * NODES_PER_BLOCK + wave * NODES_PER_WAVE;
    float* xlds = &lds[wave * NODES_PER_WAVE * ROW_STRIDE];

    // Stage W: 256 floats, one float4 per thread (L2-resident broadcast source).
    *(float4*)&wlds[tid * 4] = *(const float4*)(w + tid * 4);

    // Stage 16 rows of x per wave: 1024 float4s, 32 per lane; each load
    // instruction covers 512 contiguous bytes (perfectly coalesced).
    for (int it = 0; it < 32; ++it) {
        int f    = lane + it * 32;   // float4 index within 16x256 tile
        int row  = f >> 6;           // 64 float4 per row
        int c4   = f & 63;
        int node = node_base + row;
        float4 v = make_float4(0.0f, 0.0f, 0.0f, 0.0f);
        if (node < n) v = *(const float4*)(x + (size_t)node * D_FEAT + c4 * 4);
        *(float4*)&xlds[row * ROW_STRIDE + c4 * 4] = v;  // 16B-aligned (offsets %4==0 dwords)
    }
    __syncthreads();

    // ISA fp32 A layout (16x4): lanes 0-15 hold K=0,1 in VGPR0,1; lanes 16-31 hold K=2,3.
    // B (4x16) by the row-striping rule mirrors this: both index k + 2*(lane>>4).
    const int m     = lane & 15;
    const int khalf = (lane >> 4) * 2;
    v8f c = {0.f, 0.f, 0.f, 0.f, 0.f, 0.f, 0.f, 0.f};
#pragma unroll 8
    for (int k = 0; k < D_FEAT; k += 4) {
        v2f a  = *(const v2f*)&xlds[m * ROW_STRIDE + k + khalf];  // all 64 banks hit once
        v2f bv = *(const v2f*)&wlds[k + khalf];                   // same-address broadcast
        c = __builtin_amdgcn_wmma_f32_16x16x4_f32(
                /*neg_a=*/false, a, /*neg_b=*/false, bv,
                /*c_mod=*/(short)0, c, /*reuse_a=*/false, /*reuse_b=*/false);
    }

    // C layout: VGPR r holds row r (lanes 0-15) / row 8+r (lanes 16-31); all 16
    // columns identical (B columns identical), so lane 0 / lane 16 each own 8 nodes.
    if ((lane & 15) == 0) {
        int base = node_base + 8 * (lane >> 4);
#pragma unroll
        for (int r = 0; r < 8; ++r) {
            int node = base + r;
            if (node < n) {
                float dv = rsqrtf((float)deg[node]);  // deg >= 1 (self loop)
                dinv[node] = dv;
                hs[node]   = dv * c[r];
            }
        }
    }
}

// 4 edges per thread: two global_load_b128 (row+col streams), 4 L2-resident
// gathers, 4 native global_atomic_add_f32 (unsafeAtomicAdd -> HW fadd).
__global__ void gcn_scatter_kernel(const int* __restrict__ row, const int* __restrict__ col,
                                   const float* __restrict__ hs, float* __restrict__ acc, int e) {
    int i4 = (blockIdx.x * blockDim.x + threadIdx.x) * 4;
    if (i4 + 3 < e) {
        int4 r = *(const int4*)(row + i4);
        int4 c = *(const int4*)(col + i4);
        unsafeAtomicAdd(&acc[c.x], hs[r.x]);
        unsafeAtomicAdd(&acc[c.y], hs[r.y]);
        unsafeAtomicAdd(&acc[c.z], hs[r.z]);
        unsafeAtomicAdd(&acc[c.w], hs[r.w]);
    } else {
        for (int i = i4; i < e; ++i) unsafeAtomicAdd(&acc[col[i]], hs[row[i]]);
    }
}

__global__ void gcn_finalize_kernel(const float* __restrict__ dinv, const float* __restrict__ acc,
                                    const float* __restrict__ hs, const float* __restrict__ b,
                                    float* __restrict__ out, int n) {
    int i = blockIdx.x * blockDim.x + threadIdx.x;
    if (i < n) out[i] = dinv[i] * (acc[i] + hs[i]) + b[0];
}

extern "C" void kernel_launch(void* const* d_in, const int* in_sizes, int n_in,
                              void* d_out, int out_size, void* d_ws, size_t ws_size,
                              hipStream_t stream) {
    const float* x  = (const float*)d_in[0];
    const int*   ei = (const int*)d_in[1];
    const float* w  = (const float*)d_in[2];
    const float* b  = (const float*)d_in[3];
    float*       out = (float*)d_out;

    const int n = in_sizes[0] / D_FEAT;   // 100000
    const int e = in_sizes[1] / 2;        // 3200000
    const int* row = ei;                  // edge_index[0,:]
    const int* col = ei + e;              // edge_index[1,:]

    // Workspace: deg(int), dinv, hs, acc  -> 4 * npad * 4B ~= 1.6 MB
    const size_t npad = ((size_t)n + 255) & ~(size_t)255;
    float* wsf  = (float*)d_ws;
    int*   deg  = (int*)wsf;
    float* dinv = wsf + npad;
    float* hs   = wsf + 2 * npad;
    float* acc  = wsf + 3 * npad;

    const int e4 = (e + 3) / 4;           // threads for 4-wide edge kernels
    gcn_init_kernel<<<(n + 255) / 256, 256, 0, stream>>>(deg, acc, n);
    gcn_degree_kernel<<<(e4 + 255) / 256, 256, 0, stream>>>(col, deg, e);
    gcn_xw_wmma_kernel<<<(n + NODES_PER_BLOCK - 1) / NODES_PER_BLOCK,
                         WAVES_PER_BLOCK * 32, 0, stream>>>(x, w, deg, dinv, hs, n);
    gcn_scatter_kernel<<<(e4 + 255) / 256, 256, 0, stream>>>(row, col, hs, acc, e);
    gcn_finalize_kernel<<<(n + 255) / 256, 256, 0, stream>>>(dinv, acc, hs, b, out, n);
}